// Resample2d_4552665333943
// MI455X (gfx1250) — compile-verified
//
#include <hip/hip_runtime.h>
#include <hip/hip_bf16.h>

// Resample2d / bilinear warp, fp32, NCHW.
// B=8, C=64, H=384, W=512. Memory-bound gather: ~820 MB min traffic -> ~35us @ 23.3 TB/s.

#define RB 8
#define RC 64
#define RH 384
#define RW 512
#define RHW (RH * RW)        // 196608
#define RCHW (RC * RHW)

__global__ __launch_bounds__(256) void resample2d_warp_kernel(
    const float* __restrict__ src,   // (B, C, H, W)
    const float* __restrict__ flow,  // (B, 2, H, W)
    float* __restrict__ out)         // (B, C, H, W)
{
    const int gid = blockIdx.x * 256 + threadIdx.x;     // pixel id over B*H*W
    if (gid >= RB * RH * RW) return;

    const int w  = gid & (RW - 1);
    const int bh = gid >> 9;          // / RW
    const int h  = bh % RH;
    const int b  = bh / RH;

    // ---- flow decode (streamed once: non-temporal loads) ----
    const float* fl = flow + (size_t)b * 2 * RHW + (size_t)h * RW + w;
    const float dx = __builtin_nontemporal_load(fl);          // (b,0,h,w)
    const float dy = __builtin_nontemporal_load(fl + RHW);    // (b,1,h,w)

    const float xf  = (float)w + dx;
    const float yf  = (float)h + dy;
    const float x0f = floorf(xf);
    const float y0f = floorf(yf);
    const float alpha = xf - x0f;
    const float beta  = yf - y0f;

    const int x0i = (int)x0f;
    const int y0i = (int)y0f;
    const int x0 = min(max(x0i,     0), RW - 1);
    const int x1 = min(max(x0i + 1, 0), RW - 1);
    const int y0 = min(max(y0i,     0), RH - 1);
    const int y1 = min(max(y0i + 1, 0), RH - 1);

    const float w00 = (1.0f - alpha) * (1.0f - beta);
    const float w01 = alpha * (1.0f - beta);
    const float w10 = (1.0f - alpha) * beta;
    const float w11 = alpha * beta;

    // ---- per-pixel tap base pointers (c = 0), channel stride = RHW ----
    const size_t bbase = (size_t)b * RCHW;
    const float* p00 = src + bbase + (size_t)y0 * RW + x0;
    const float* p01 = src + bbase + (size_t)y0 * RW + x1;
    const float* p10 = src + bbase + (size_t)y1 * RW + x0;
    const float* p11 = src + bbase + (size_t)y1 * RW + x1;
    float*       po  = out + bbase + (size_t)h  * RW + w;

    // Unroll 8 channels per trip: inner byte offsets u*RHW*4 <= 5.25 MB fit the
    // signed 24-bit global-load immediate; 32 loads in flight per trip.
    #pragma unroll 1
    for (int c = 0; c < RC; c += 8) {
        float v00[8], v01[8], v10[8], v11[8];
        #pragma unroll
        for (int u = 0; u < 8; ++u) {
            v00[u] = p00[u * RHW];
            v01[u] = p01[u * RHW];
            v10[u] = p10[u * RHW];
            v11[u] = p11[u * RHW];
        }
        #pragma unroll
        for (int u = 0; u < 8; ++u) {
            float r = fmaf(w00, v00[u],
                      fmaf(w01, v01[u],
                      fmaf(w10, v10[u],
                           w11 * v11[u])));
            // write-once output: non-temporal store keeps L2 for the gather reuse
            __builtin_nontemporal_store(r, po + u * RHW);
        }
        p00 += 8 * RHW; p01 += 8 * RHW; p10 += 8 * RHW; p11 += 8 * RHW;
        po  += 8 * RHW;
    }
}

extern "C" void kernel_launch(void* const* d_in, const int* in_sizes, int n_in,
                              void* d_out, int out_size, void* d_ws, size_t ws_size,
                              hipStream_t stream) {
    const float* src  = (const float*)d_in[0];   // (B, C, H, W) fp32
    const float* flow = (const float*)d_in[1];   // (B, 2, H, W) fp32
    float* out = (float*)d_out;                  // (B, C, H, W) fp32

    const int npix = RB * RH * RW;               // 1,572,864
    const int blocks = (npix + 255) / 256;       // 6144
    resample2d_warp_kernel<<<blocks, 256, 0, stream>>>(src, flow, out);
}